// SAGE_472446402719
// MI455X (gfx1250) — compile-verified
//
#include <hip/hip_runtime.h>

typedef __attribute__((ext_vector_type(2))) float v2f;
typedef __attribute__((ext_vector_type(8))) float v8f;
typedef __attribute__((ext_vector_type(4))) unsigned int v4u;
typedef __attribute__((ext_vector_type(8))) int v8i;
typedef __attribute__((ext_vector_type(4))) int v4i;

constexpr int IN  = 128;
constexpr int HID = 16;
constexpr int OUT = 64;

// ---------------------------------------------------------------- zero
__global__ void zero_kernel(float* __restrict__ p, int n) {
    int i = blockIdx.x * blockDim.x + threadIdx.x;
    if (i < n) p[i] = 0.0f;
}

// ---------------------------------------------------------------- degree
__global__ void degree_kernel(const int* __restrict__ dst, float* __restrict__ deg, int E) {
    int e = blockIdx.x * blockDim.x + threadIdx.x;
    if (e < E) atomicAdd(&deg[dst[e]], 1.0f);
}

// ---------------------------------------------------------------- GEMM1
// P = x @ [W_self1 | W_neigh1]   (M=50000, K=128, N=16+16)
// One wave per 16-row tile. x tile brought into LDS by the Tensor Data Mover
// (tensor_load_to_lds, 2D descriptor), then consumed via ds_load_b64 A-fragments.
// f32 WMMA 16x16x4: A 16x4 (2 VGPR), B 4x16 (2 VGPR), C/D 16x16 (8 VGPR).
__global__ void __launch_bounds__(32) gemm1_kernel(
    const float* __restrict__ x,
    const float* __restrict__ Wself,    // [128,16]
    const float* __restrict__ Wneigh,   // [128,16]
    float* __restrict__ Pself,          // [N,16]
    float* __restrict__ Pneigh,         // [N,16]
    int nrows)
{
    __shared__ float tile[16 * IN];     // 8 KB

    const int lane = threadIdx.x;
    const int row0 = blockIdx.x * 16;
    const int rsel = lane >> 4;         // 0: lanes 0-15, 1: lanes 16-31
    const int mrow = lane & 15;

    // ---- TDM: DMA the 16x128 f32 tile (rows row0..row0+15 of x) into LDS ----
    // D# group 0: {count=1 | flags}, lds_addr, global_addr[31:0],
    //             global_addr[56:32] | type=2 in bits[31:30]
    const unsigned lds_off = (unsigned)(size_t)(void*)tile;           // LDS aperture: low 32 bits
    const unsigned long long ga =
        (unsigned long long)(const void*)(x + (size_t)row0 * IN);
    v4u g0 = { 1u, lds_off, (unsigned)ga, (unsigned)(ga >> 32) | (2u << 30) };

    // D# group 1: data_size=4B; tensor_dim0=128; tensor_dim1=nrows;
    //             tile_dim0=128; tile_dim1=16; tensor_dim0_stride=128
    const unsigned td0 = (unsigned)IN;
    const unsigned td1 = (unsigned)nrows;
    v8i g1;
    g1[0] = (int)(2u << 16);                                   // data_size = 2 -> 4 bytes
    g1[1] = (int)(td0 << 16);                                  // tensor_dim0[15:0]
    g1[2] = (int)((td0 >> 16) | ((td1 & 0xFFFFu) << 16));      // dim0 hi | dim1 lo
    g1[3] = (int)(((td1 >> 16) & 0xFFFFu) | ((unsigned)IN << 16)); // dim1 hi | tile_dim0=128
    g1[4] = 16;                                                // tile_dim1=16, tile_dim2=0
    g1[5] = (int)td0;                                          // tensor_dim0_stride lo
    g1[6] = 0;                                                 // stride hi | dim1_stride lo (unused, 2D tile)
    g1[7] = 0;
    v4i gz4 = { 0, 0, 0, 0 };                                  // groups 2/3 unused (<=2D)
    v8i gz8 = { 0, 0, 0, 0, 0, 0, 0, 0 };
    __builtin_amdgcn_tensor_load_to_lds(g0, g1, gz4, gz4, gz8, 0);
    __builtin_amdgcn_s_wait_tensorcnt(0);
    __syncthreads();

    v8f cs = {};
    v8f cn = {};

#pragma unroll
    for (int k0 = 0; k0 < IN; k0 += 4) {
        const int kA = k0 + (rsel << 1);            // lanes<16: k0,k0+1 ; lanes>=16: k0+2,k0+3
        // A fragment: row mrow, contiguous K pair -> ds_load_b64
        v2f a = *(const v2f*)&tile[mrow * IN + kA];
        // B fragments: W[kA][mrow], W[kA+1][mrow]   (row stride HID)
        v2f bs = { Wself [kA * HID + mrow], Wself [(kA + 1) * HID + mrow] };
        v2f bn = { Wneigh[kA * HID + mrow], Wneigh[(kA + 1) * HID + mrow] };
        cs = __builtin_amdgcn_wmma_f32_16x16x4_f32(false, a, false, bs, (short)0, cs, false, false);
        cn = __builtin_amdgcn_wmma_f32_16x16x4_f32(false, a, false, bn, (short)0, cn, false, false);
    }

    // D layout: VGPR i -> row (i + 8*rsel), col = mrow
#pragma unroll
    for (int i = 0; i < 8; ++i) {
        const size_t r = (size_t)(row0 + i + (rsel << 3));
        Pself [r * HID + mrow] = cs[i];
        Pneigh[r * HID + mrow] = cn[i];
    }
}

// ---------------------------------------------------------------- edge scatter
// agg[dst] += feat[src]   (16 floats / edge). 4 threads per edge, float4 each.
__global__ void scatter_kernel(const float* __restrict__ feat,
                               const int* __restrict__ src,
                               const int* __restrict__ dst,
                               float* __restrict__ agg, int E) {
    int t = blockIdx.x * blockDim.x + threadIdx.x;
    int e = t >> 2;
    if (e >= E) return;
    int c = (t & 3) << 2;
    int s = src[e], d = dst[e];
    __builtin_prefetch(&feat[(size_t)s * HID + c], 0, 1);   // global_prefetch_b8
    float4 v = *(const float4*)&feat[(size_t)s * HID + c];
    float* o = &agg[(size_t)d * HID + c];
    atomicAdd(o + 0, v.x);
    atomicAdd(o + 1, v.y);
    atomicAdd(o + 2, v.z);
    atomicAdd(o + 3, v.w);
}

// ---------------------------------------------------------------- layer-1 epilogue
// h = relu(Pself + agg/deg + b1)
__global__ void combine1_kernel(const float* __restrict__ Pself,
                                const float* __restrict__ agg,
                                const float* __restrict__ deg,
                                const float* __restrict__ b1,
                                float* __restrict__ h, int total) {
    int i = blockIdx.x * blockDim.x + threadIdx.x;
    if (i >= total) return;
    int node = i >> 4;           // /HID
    int j    = i & 15;           // %HID
    float dv = deg[node];
    dv = dv < 1.0f ? 1.0f : dv;
    float v = Pself[i] + agg[i] / dv + b1[j];
    h[i] = v > 0.0f ? v : 0.0f;
}

// ---------------------------------------------------------------- GEMM2
// out = [h | aggh/deg] @ [W_self2 ; W_neigh2] + b2   (M=50000, K=32, N=64)
__global__ void __launch_bounds__(32) gemm2_kernel(
    const float* __restrict__ h,       // [N,16]
    const float* __restrict__ aggh,    // [N,16]  (unnormalized sums)
    const float* __restrict__ deg,
    const float* __restrict__ Wself,   // [16,64]
    const float* __restrict__ Wneigh,  // [16,64]
    const float* __restrict__ b2,      // [64]
    float* __restrict__ out)           // [N,64]
{
    const int lane = threadIdx.x;
    const int row0 = blockIdx.x * 16;
    const int rsel = lane >> 4;
    const int mrow = lane & 15;
    const int r    = row0 + mrow;

    float dv = deg[r];
    dv = dv < 1.0f ? 1.0f : dv;
    const float inv = 1.0f / dv;

    // A fragments for all 8 K-steps: k = 4s + 2*rsel, pair (k,k+1).
    // s=0..3 -> h columns, s=4..7 -> normalized aggh columns.
    v2f afrag[8];
#pragma unroll
    for (int s = 0; s < 4; ++s) {
        const int kA = 4 * s + (rsel << 1);
        afrag[s] = *(const v2f*)&h[(size_t)r * HID + kA];
        v2f t = *(const v2f*)&aggh[(size_t)r * HID + kA];
        afrag[s + 4] = t * inv;
    }

    // init accumulators with bias (row-constant per N-tile column)
    v8f c[4];
#pragma unroll
    for (int n = 0; n < 4; ++n) {
        const float bb = b2[16 * n + mrow];
#pragma unroll
        for (int i = 0; i < 8; ++i) c[n][i] = bb;
    }

#pragma unroll
    for (int s = 0; s < 8; ++s) {
        const int k = 4 * s + (rsel << 1);   // even; (k,k+1) never straddles k=16
        const float* W = (k < 16) ? (Wself + (size_t)k * OUT)
                                  : (Wneigh + (size_t)(k - 16) * OUT);
#pragma unroll
        for (int n = 0; n < 4; ++n) {
            v2f b = { W[16 * n + mrow], W[OUT + 16 * n + mrow] };
            c[n] = __builtin_amdgcn_wmma_f32_16x16x4_f32(false, afrag[s], false, b,
                                                         (short)0, c[n], false, false);
        }
    }

#pragma unroll
    for (int n = 0; n < 4; ++n) {
#pragma unroll
        for (int i = 0; i < 8; ++i) {
            out[(size_t)(row0 + i + (rsel << 3)) * OUT + 16 * n + mrow] = c[n][i];
        }
    }
}

// ---------------------------------------------------------------- host
extern "C" void kernel_launch(void* const* d_in, const int* in_sizes, int n_in,
                              void* d_out, int out_size, void* d_ws, size_t ws_size,
                              hipStream_t stream) {
    const float* x    = (const float*)d_in[0];
    const int*   src  = (const int*)d_in[1];
    const int*   dst  = (const int*)d_in[2];
    const float* Ws1  = (const float*)d_in[3];
    const float* Wn1  = (const float*)d_in[4];
    const float* b1   = (const float*)d_in[5];
    const float* Ws2  = (const float*)d_in[6];
    const float* Wn2  = (const float*)d_in[7];
    const float* b2   = (const float*)d_in[8];
    float*       out  = (float*)d_out;

    const int N = in_sizes[0] / IN;   // 50000 (divisible by 16)
    const int E = in_sizes[1];        // 800000

    // workspace layout: [deg | agg1 | aggh | Pself | Pneigh | h]
    float* deg    = (float*)d_ws;
    float* agg1   = deg    + N;
    float* aggh   = agg1   + (size_t)N * HID;
    float* Pself  = aggh   + (size_t)N * HID;
    float* Pneigh = Pself  + (size_t)N * HID;
    float* h      = Pneigh + (size_t)N * HID;

    const int zn = N * (1 + 2 * HID);  // deg + agg1 + aggh are contiguous
    zero_kernel<<<(zn + 255) / 256, 256, 0, stream>>>(deg, zn);

    degree_kernel<<<(E + 255) / 256, 256, 0, stream>>>(dst, deg, E);

    gemm1_kernel<<<N / 16, 32, 0, stream>>>(x, Ws1, Wn1, Pself, Pneigh, N);

    scatter_kernel<<<(4 * E + 255) / 256, 256, 0, stream>>>(Pneigh, src, dst, agg1, E);

    combine1_kernel<<<(N * HID + 255) / 256, 256, 0, stream>>>(Pself, agg1, deg, b1, h, N * HID);

    scatter_kernel<<<(4 * E + 255) / 256, 256, 0, stream>>>(h, src, dst, aggh, E);

    gemm2_kernel<<<N / 16, 32, 0, stream>>>(h, aggh, deg, Ws2, Wn2, b2, out);
}